// EMD_37349035606526
// MI455X (gfx1250) — compile-verified
//
#include <hip/hip_runtime.h>

typedef float v2f __attribute__((ext_vector_type(2)));
typedef float v8f __attribute__((ext_vector_type(8)));

#define HW   65536
#define NB   8
#define BINS 256
#define SLICES 32
#define PIX_PER_BLOCK (HW / SLICES)          // 2048 pixels per block
#define V4_PER_THREAD (PIX_PER_BLOCK / 256 / 4) // 2 float4 loads per thread/input
#define XPITCH 258                            // padded LDS row pitch (bank-conflict-free)

// Gaussian splat: with sigma=0.1 and bin spacing 1.0, exp(-50*d^2) is exact
// fp32 zero for |d| >= 1.44, so only bins j-1, j, j+1 ever contribute.
__device__ __forceinline__ void splat3(float v, float* hrow) {
    int j = __float2int_rn(v);
#pragma unroll
    for (int dj = -1; dj <= 1; ++dj) {
        int k = j + dj;
        if (k >= 0 && k < BINS) {
            float d = v - (float)k;
            float w = __expf(-50.0f * d * d);
            atomicAdd(&hrow[k], w);          // ds_add_f32 (LDS float atomic)
        }
    }
}

__global__ void emd_hist_kernel(const float4* __restrict__ in1,
                                const float4* __restrict__ in2,
                                float* __restrict__ hist) {
    __shared__ float h[2][BINS];
    const int t = threadIdx.x;               // 256 threads
    h[0][t] = 0.0f;
    h[1][t] = 0.0f;
    __syncthreads();

    const int batch = blockIdx.y;
    const int slice = blockIdx.x;
    // base in float4 units
    const int base4 = (batch * HW + slice * PIX_PER_BLOCK) >> 2;

#pragma unroll
    for (int i = 0; i < V4_PER_THREAD; ++i) {
        const int idx = base4 + i * 256 + t; // coalesced global_load_b128
        float4 p1 = in1[idx];
        float4 p2 = in2[idx];
        splat3(p1.x * 255.0f, h[0]);
        splat3(p1.y * 255.0f, h[0]);
        splat3(p1.z * 255.0f, h[0]);
        splat3(p1.w * 255.0f, h[0]);
        splat3(p2.x * 255.0f, h[1]);
        splat3(p2.y * 255.0f, h[1]);
        splat3(p2.z * 255.0f, h[1]);
        splat3(p2.w * 255.0f, h[1]);
    }
    __syncthreads();

    // merge block-private histograms into global scratch
    atomicAdd(&hist[(0 * NB + batch) * BINS + t], h[0][t]); // global_atomic_add_f32
    atomicAdd(&hist[(1 * NB + batch) * BINS + t], h[1][t]);
}

// Finalize: normalize pdfs, diff, then cumsum via lower-triangular-ones matmul
// on V_WMMA_F32_16X16X4_F32 (exact fp32), then mean of squares per batch.
__global__ void emd_final_kernel(const float* __restrict__ hist,
                                 float* __restrict__ out) {
    __shared__ float Xt[16][XPITCH]; // diff matrix, batch-major: [batch][bin]
    __shared__ float red[BINS][16];  // reduction scratch for the 16 bin-sums
    __shared__ float scale[16];
    __shared__ float emdacc[16];

    const int t = threadIdx.x;       // 256 threads = 8 waves (wave32)

    float r1[NB], r2[NB];
#pragma unroll
    for (int b = 0; b < NB; ++b) {
        r1[b] = hist[b * BINS + t];
        r2[b] = hist[(NB + b) * BINS + t];
        red[t][b]      = r1[b];
        red[t][NB + b] = r2[b];
    }
    if (t < 16) emdacc[t] = 0.0f;
    __syncthreads();

    // tree-reduce the 16 columns over the 256 bins
    for (int s = 128; s > 0; s >>= 1) {
        if (t < s) {
#pragma unroll
            for (int p = 0; p < 16; ++p) red[t][p] += red[t + s][p];
        }
        __syncthreads();
    }
    // pdf = h/HW; norm = sum/HW + eps  =>  pdf_n = h / (sum + HW*eps)
    if (t < 16) scale[t] = 1.0f / (red[0][t] + (float)HW * 1e-10f);
    __syncthreads();

#pragma unroll
    for (int b = 0; b < NB; ++b)
        Xt[b][t] = r1[b] * scale[b] - r2[b] * scale[NB + b];
#pragma unroll
    for (int b = NB; b < 16; ++b) Xt[b][t] = 0.0f;
    __syncthreads();

    // cdf_diff(256x8) = L(256x256, lower-tri ones) * X(256x8)
    const int wave = t >> 5;
    const int lane = t & 31;
    const int mn   = lane & 15;  // A: row within tile / B,D: column (batch)
    const int half = lane >> 4;  // selects K pair {0,2} vs rows +8 in D

    for (int mt = wave; mt < 16; mt += 8) {       // 16 M-tiles over 8 waves
        const int mrow = mt * 16 + mn;
        v8f c = {0.f, 0.f, 0.f, 0.f, 0.f, 0.f, 0.f, 0.f};
#pragma unroll 1                                  // keep rolled: no VGPR spill
        for (int kt = 0; kt < 64; ++kt) {         // K = 256 in steps of 4
            const int k0 = kt * 4 + half * 2;
            v2f a, bb;
            // A tile of L generated in registers: L[m][k] = (k <= m)
            a.x = (k0     <= mrow) ? 1.0f : 0.0f; // VGPR0: K = {0,2}
            a.y = (k0 + 1 <= mrow) ? 1.0f : 0.0f; // VGPR1: K = {1,3}
            // contiguous, conflict-free ds_load_b64: B[k][n] = Xt[n][k]
            bb = *(const v2f*)&Xt[mn][k0];
            c = __builtin_amdgcn_wmma_f32_16x16x4_f32(
                    false, a, false, bb, (short)0, c, false, false);
        }
        // D layout: VGPR r holds row (mt*16 + r + 8*half), column mn
        if (mn < NB) {
            float acc = 0.0f;
#pragma unroll
            for (int r = 0; r < 8; ++r) acc += c[r] * c[r];
            atomicAdd(&emdacc[mn], acc);
        }
    }
    __syncthreads();
    if (t < NB) out[t] = emdacc[t] * (1.0f / 256.0f); // mean over 256 bins
}

extern "C" void kernel_launch(void* const* d_in, const int* in_sizes, int n_in,
                              void* d_out, int out_size, void* d_ws, size_t ws_size,
                              hipStream_t stream) {
    (void)in_sizes; (void)n_in; (void)out_size; (void)ws_size;
    const float4* in1 = (const float4*)d_in[0];
    const float4* in2 = (const float4*)d_in[1];
    // d_in[2] (bins) is exactly 0..255 step 1.0 by construction; folded into math.
    float* hist = (float*)d_ws;                  // 2*8*256 floats = 16 KB

    hipMemsetAsync(hist, 0, 2 * NB * BINS * sizeof(float), stream);
    emd_hist_kernel<<<dim3(SLICES, NB), 256, 0, stream>>>(in1, in2, hist);
    emd_final_kernel<<<1, 256, 0, stream>>>(hist, (float*)d_out);
}